// Embed_8297876816108
// MI455X (gfx1250) — compile-verified
//
#include <hip/hip_runtime.h>
#include <stdint.h>

// Embedding gather: out[row, :] = w[ids[row], :]
// rows = 8*2048 = 16384, UNITS = 512 floats (2048 bytes per row).
// Pure bandwidth problem (~64 MB total @ 23.3 TB/s ~ 2.8 us).
// Implemented with the CDNA5 async global<->LDS data-mover path:
//   global_load_async_to_lds_b128  (memory -> LDS, ASYNCcnt)
//   s_wait_asynccnt 0
//   global_store_async_from_lds_b128 (LDS -> memory, ASYNCcnt)
// One wave32 per row; each lane owns 16 bytes; 4 async ops of 512 B each
// cover the 2048-byte row (INST_OFFSET is added to both LDS and global
// addresses per the ISA, so one base address pair serves all four).

#define UNITS            512
#define ROW_BYTES        (UNITS * 4)     // 2048
#define WAVES_PER_BLOCK  8
#define BLOCK_THREADS    (WAVES_PER_BLOCK * 32)

__global__ __launch_bounds__(BLOCK_THREADS)
void embed_gather_async_kernel(const int* __restrict__ ids,
                               const float* __restrict__ w,
                               float* __restrict__ out,
                               int n_rows) {
    __shared__ char smem[WAVES_PER_BLOCK * ROW_BYTES];   // 16 KB

    const int lane = threadIdx.x & 31;
    const int wave = threadIdx.x >> 5;
    const int row  = blockIdx.x * WAVES_PER_BLOCK + wave;
    if (row >= n_rows) return;   // wave-uniform; EXEC stays all-ones for async ops

    const int id = ids[row];

    // Per-lane 16-byte slice of the source row.
    const char* gsrc = (const char*)(w + (size_t)id * UNITS) + lane * 16;
    // LDS byte offset: low 32 bits of the flat shared address (flat->LDS
    // addressing truncates to the workgroup-relative LDS offset).
    unsigned ldsOff =
        (unsigned)(uintptr_t)(&smem[wave * ROW_BYTES + lane * 16]);

    // Memory -> LDS, 512 B per instruction, 4 instructions = full row.
    asm volatile("global_load_async_to_lds_b128 %0, %1, off"
                 :: "v"(ldsOff), "v"(gsrc) : "memory");
    asm volatile("global_load_async_to_lds_b128 %0, %1, off offset:512"
                 :: "v"(ldsOff), "v"(gsrc) : "memory");
    asm volatile("global_load_async_to_lds_b128 %0, %1, off offset:1024"
                 :: "v"(ldsOff), "v"(gsrc) : "memory");
    asm volatile("global_load_async_to_lds_b128 %0, %1, off offset:1536"
                 :: "v"(ldsOff), "v"(gsrc) : "memory");

    // Wait for this wave's async loads to land in LDS.
    asm volatile("s_wait_asynccnt 0" ::: "memory");

    // LDS -> memory for the destination row. Outstanding async stores are
    // drained by S_ENDPGM's implicit wait-idle.
    char* gdst = (char*)(out + (size_t)row * UNITS) + lane * 16;
    asm volatile("global_store_async_from_lds_b128 %0, %1, off"
                 :: "v"(gdst), "v"(ldsOff) : "memory");
    asm volatile("global_store_async_from_lds_b128 %0, %1, off offset:512"
                 :: "v"(gdst), "v"(ldsOff) : "memory");
    asm volatile("global_store_async_from_lds_b128 %0, %1, off offset:1024"
                 :: "v"(gdst), "v"(ldsOff) : "memory");
    asm volatile("global_store_async_from_lds_b128 %0, %1, off offset:1536"
                 :: "v"(gdst), "v"(ldsOff) : "memory");
}

extern "C" void kernel_launch(void* const* d_in, const int* in_sizes, int n_in,
                              void* d_out, int out_size, void* d_ws, size_t ws_size,
                              hipStream_t stream) {
    const int*   ids = (const int*)d_in[0];    // [8, 2048] int32
    const float* w   = (const float*)d_in[1];  // [50257, 512] f32
    float*       out = (float*)d_out;          // [8*2048, 512] f32

    const int n_rows = in_sizes[0];            // 16384
    const int grid   = (n_rows + WAVES_PER_BLOCK - 1) / WAVES_PER_BLOCK;

    hipLaunchKernelGGL(embed_gather_async_kernel,
                       dim3(grid), dim3(BLOCK_THREADS), 0, stream,
                       ids, w, out, n_rows);
}